// XEyTransformerLayer_78116865179719
// MI455X (gfx1250) — compile-verified
//
#include <hip/hip_runtime.h>
#include <math.h>

typedef float v2f __attribute__((ext_vector_type(2)));
typedef float v8f __attribute__((ext_vector_type(8)));

#define DXc 256
#define DEc 64
#define Nc  256
#define FFXc 2048
#define FFEc 128
#define EPSc 1e-5f

__device__ __forceinline__ v8f vzero8() {
  v8f z; for (int i = 0; i < 8; ++i) z[i] = 0.f; return z;
}

// A fragment: 16x4 fp32 tile, p points at (row0,k0), row stride ld.
// lane&15 -> M ; VGPR v + 2*(lane>>4) -> K
__device__ __forceinline__ v2f ldA(const float* p, int ld, int lane) {
  const float* q = p + (lane & 15) * ld + ((lane >> 4) << 1);
  v2f r; r.x = q[0]; r.y = q[1]; return r;
}
// B fragment: 4x16 fp32 tile, p points at (k0,n0), row stride ld.
__device__ __forceinline__ v2f ldB(const float* p, int ld, int lane) {
  const float* q = p + (((lane >> 4) << 1)) * ld + (lane & 15);
  v2f r; r.x = q[0]; r.y = q[ld]; return r;
}

__device__ __forceinline__ v8f wmma4(v2f a, v2f b, v8f c) {
  return __builtin_amdgcn_wmma_f32_16x16x4_f32(false, a, false, b, (short)0, c,
                                               false, false);
}

// ---------------------------------------------------------------------------
// Tensor Data Mover: 1-D contiguous global->LDS DMA (clang-23 6-arg builtin).
// ---------------------------------------------------------------------------
#if __has_builtin(__builtin_amdgcn_tensor_load_to_lds)
#define HAVE_TDM 1
typedef unsigned int u32x4 __attribute__((ext_vector_type(4)));
typedef int i32x4 __attribute__((ext_vector_type(4)));
typedef int i32x8 __attribute__((ext_vector_type(8)));

__device__ __forceinline__ void tdm_load_1d(unsigned lds_off, const void* gaddr,
                                            unsigned nelem /* f32 elements */) {
  unsigned long long ga = (unsigned long long)(uintptr_t)gaddr;
  u32x4 g0;
  g0[0] = 1u;                                      // count=1 (valid), user mode
  g0[1] = lds_off;                                 // LDS byte address
  g0[2] = (unsigned)(ga & 0xFFFFFFFFu);            // global_addr[31:0]
  g0[3] = (unsigned)((ga >> 32) & 0x01FFFFFFu)     // global_addr[56:32]
          | (2u << 30);                            // type = 2 ("image")
  i32x8 g1;
  g1[0] = (int)(2u << 16);                         // wg_mask=0, data_size=4B
  g1[1] = 0;                                       // tensor_dim0[15:0] = 0
  g1[2] = (int)(((1u << 30) >> 16) & 0xFFFFu)      // tensor_dim0 = 1<<30 (hi16)
          | (int)(1u << 16);                       // tensor_dim1 = 1 (lo16)
  g1[3] = (int)((nelem & 0xFFFFu) << 16);          // tile_dim0 = nelem
  g1[4] = (int)1;                                  // tile_dim1 = 1, tile_dim2 = 0
  g1[5] = (int)nelem;                              // tensor_dim0_stride lo32
  g1[6] = 0;
  g1[7] = 0;
  i32x4 z4; z4[0] = 0; z4[1] = 0; z4[2] = 0; z4[3] = 0;  // groups 2/3 (<=2D)
  i32x8 z8; for (int q = 0; q < 8; ++q) z8[q] = 0;
  __builtin_amdgcn_tensor_load_to_lds(g0, g1, z4, z4, z8, 0);
}
#else
#define HAVE_TDM 0
#endif

// ---------------------------------------------------------------------------
// Kernel 1: Q/K/V = (X @ W + b) * xm  (three 1024x256x256 GEMMs)
// 4-wide N blocking: 1 A-frag feeds 4 WMMAs. 768 wave-tasks, 8 waves/block.
// ---------------------------------------------------------------------------
__global__ __launch_bounds__(256) void qkv_kernel(
    const float* __restrict__ X,
    const float* __restrict__ qW, const float* __restrict__ qB,
    const float* __restrict__ kW, const float* __restrict__ kB,
    const float* __restrict__ vW, const float* __restrict__ vB,
    const float* __restrict__ mask,
    float* __restrict__ Q, float* __restrict__ K, float* __restrict__ V) {
  int lane = threadIdx.x & 31;
  int gw = blockIdx.x * (blockDim.x >> 5) + (threadIdx.x >> 5);  // 0..767
  int which = gw >> 8;             // 0=Q,1=K,2=V (256 tasks each)
  int t = gw & 255;
  int mt = t >> 2, ng = t & 3;     // n0 = ng*64
  const float* W = which == 0 ? qW : (which == 1 ? kW : vW);
  const float* B = which == 0 ? qB : (which == 1 ? kB : vB);
  float* O = which == 0 ? Q : (which == 1 ? K : V);

  v8f acc[4];
  acc[0] = vzero8(); acc[1] = vzero8(); acc[2] = vzero8(); acc[3] = vzero8();
  const float* Ab = X + (size_t)mt * 16 * DXc;
  const float* Bb = W + (ng << 6);
  for (int k = 0; k < DXc; k += 4) {
    v2f a = ldA(Ab + k, DXc, lane);
#pragma unroll
    for (int nn = 0; nn < 4; ++nn)
      acc[nn] = wmma4(a, ldB(Bb + (size_t)k * DXc + nn * 16, DXc, lane), acc[nn]);
  }
  int row0 = (lane >> 4) << 3, col = lane & 15;
#pragma unroll
  for (int nn = 0; nn < 4; ++nn) {
    float bias = B[(ng << 6) + nn * 16 + col];
    for (int r = 0; r < 8; ++r) {
      int m = mt * 16 + row0 + r;
      O[(size_t)m * DXc + (ng << 6) + nn * 16 + col] = (acc[nn][r] + bias) * mask[m];
    }
  }
}

// ---------------------------------------------------------------------------
// Kernel 2: fused edge path, one workgroup (16 waves) per (b,i).
//  - E tile double-buffered into LDS by the Tensor Data Mover
//  - emul/eadd B-fragments hoisted into registers across the j loop
//  - online softmax per channel, Y staged via LDS into the eout WMMA GEMM
// ---------------------------------------------------------------------------
__global__ __launch_bounds__(512) void edge_fused_kernel(
    const float* __restrict__ E, const float* __restrict__ mask,
    const float* __restrict__ Qb, const float* __restrict__ Kb,
    const float* __restrict__ Vb,
    const float* __restrict__ emulW, const float* __restrict__ emulB,
    const float* __restrict__ eaddW, const float* __restrict__ eaddB,
    const float* __restrict__ eoutW, const float* __restrict__ eoutB,
    float* __restrict__ newE, float* __restrict__ wV) {
  __shared__ float Ylds[16 * DXc];
  __shared__ __align__(16) float Ebuf[2][16 * DEc];
  int lane = threadIdx.x & 31;
  int wv = threadIdx.x >> 5;            // 0..15
  int brow = blockIdx.x;                // b*256 + i
  int b = brow >> 8;
  int c = (wv << 4) + (lane & 15);
  int rbase = (lane >> 4) << 3;
  int col = lane & 15;

  float qv  = Qb[(size_t)brow * DXc + c];
  float xmi = mask[brow];
  const float invs = 0.17677669529663688f;  // 1/sqrt(32)

  // Hoist the two weight panels (64x16 each) into registers: reused 16x.
  v2f bf1[16], bf2[16];
#pragma unroll
  for (int kk = 0; kk < 16; ++kk) {
    bf1[kk] = ldB(emulW + (size_t)(kk * 4) * DXc + (wv << 4), DXc, lane);
    bf2[kk] = ldB(eaddW + (size_t)(kk * 4) * DXc + (wv << 4), DXc, lane);
  }
  float b1v = emulB[c], b2v = eaddB[c];

  float m_run = -__builtin_huge_valf(), s_run = 0.f, acc = 0.f;

#if HAVE_TDM
  unsigned ldsb[2];
  ldsb[0] = (unsigned)(uintptr_t)&Ebuf[0][0];
  ldsb[1] = (unsigned)(uintptr_t)&Ebuf[1][0];
  if (wv == 0)   // prefetch tile j0=0
    tdm_load_1d(ldsb[0], E + (size_t)brow * Nc * DEc, 16 * DEc);
#endif

  for (int j0 = 0; j0 < Nc; j0 += 16) {
    int cur = (j0 >> 4) & 1;
#if HAVE_TDM
    if (wv == 0) {
      if (j0 < Nc - 16) {
        tdm_load_1d(ldsb[cur ^ 1],
                    E + ((size_t)brow * Nc + j0 + 16) * DEc, 16 * DEc);
        __builtin_amdgcn_s_wait_tensorcnt(1);   // tile j0 complete (in-order)
      } else {
        __builtin_amdgcn_s_wait_tensorcnt(0);
      }
    }
    __syncthreads();                            // publish Ebuf[cur]
    const float* Ab = &Ebuf[cur][0];
#else
    for (int idx = threadIdx.x; idx < 16 * DEc; idx += 512)
      Ebuf[0][idx] = E[((size_t)brow * Nc + j0) * DEc + idx];
    __syncthreads();
    const float* Ab = &Ebuf[0][0];
#endif

    // --- E1/E2 projections for this wave's 16 channels
    v8f e1 = vzero8(), e2 = vzero8();
#pragma unroll
    for (int kk = 0; kk < 16; ++kk) {
      v2f a = ldA(Ab + kk * 4, DEc, lane);
      e1 = wmma4(a, bf1[kk], e1);
      e2 = wmma4(a, bf2[kk], e2);
    }

    // --- Y + online softmax state update
    float p[8];
    float mt = -__builtin_huge_valf();
    for (int r = 0; r < 8; ++r) {
      int jl = rbase + r;
      int j = j0 + jl;
      float xmj = mask[(b << 8) + j];
      float em = xmi * xmj;
      float kv = Kb[((size_t)(b << 8) + j) * DXc + c];
      float E1v = (e1[r] + b1v) * em;
      float E2v = (e2[r] + b2v) * em;
      float yv = qv * kv * invs * (E1v + 1.f) + E2v;
      Ylds[jl * DXc + c] = yv;
      float ym = (xmj > 0.f) ? yv : -1e9f;
      p[r] = ym;
      mt = fmaxf(mt, ym);
    }
    mt = fmaxf(mt, __shfl_xor(mt, 16, 32));
    float m_new = fmaxf(m_run, mt);
    float scale = __expf(m_run - m_new);
    float tsum = 0.f, tacc = 0.f;
    for (int r = 0; r < 8; ++r) {
      int j = j0 + rbase + r;
      float pe = __expf(p[r] - m_new);
      tsum += pe;
      tacc += pe * Vb[((size_t)(b << 8) + j) * DXc + c];
    }
    tsum += __shfl_xor(tsum, 16, 32);
    tacc += __shfl_xor(tacc, 16, 32);
    s_run = s_run * scale + tsum;
    acc   = acc   * scale + tacc;
    m_run = m_new;

    __syncthreads();                            // Ylds fully written

    // --- newE_pre = (Y(16x256) @ eout_W(256x64) + b) * em  (waves 0..3)
    if (wv < 4) {
      int n0 = wv << 4;
      v8f a8 = vzero8();
      for (int k = 0; k < DXc; k += 4)
        a8 = wmma4(ldA(Ylds + k, DXc, lane),
                   ldB(eoutW + (size_t)k * DEc + n0, DEc, lane), a8);
      float bb = eoutB[n0 + col];
      for (int r = 0; r < 8; ++r) {
        int jl = rbase + r;
        int j = j0 + jl;
        float em = xmi * mask[(b << 8) + j];
        newE[((size_t)brow * Nc + j) * DEc + n0 + col] = (a8[r] + bb) * em;
      }
    }
    __syncthreads();                            // Ylds/Ebuf free for next tile
  }

  if (lane < 16)
    wV[(size_t)brow * DXc + c] = acc / s_run;
}

// ---------------------------------------------------------------------------
// Kernel 3: pre1 = X + (wV @ xout_W + b) * xm   (4-wide N blocking)
// ---------------------------------------------------------------------------
__global__ __launch_bounds__(256) void xout_kernel(
    const float* __restrict__ wV, const float* __restrict__ W,
    const float* __restrict__ B, const float* __restrict__ mask,
    const float* __restrict__ X, float* __restrict__ pre1) {
  int lane = threadIdx.x & 31;
  int gw = blockIdx.x * (blockDim.x >> 5) + (threadIdx.x >> 5);  // 0..255
  int mt = gw >> 2, ng = gw & 3;
  v8f acc[4];
  acc[0] = vzero8(); acc[1] = vzero8(); acc[2] = vzero8(); acc[3] = vzero8();
  const float* Ab = wV + (size_t)mt * 16 * DXc;
  const float* Bb = W + (ng << 6);
  for (int k = 0; k < DXc; k += 4) {
    v2f a = ldA(Ab + k, DXc, lane);
#pragma unroll
    for (int nn = 0; nn < 4; ++nn)
      acc[nn] = wmma4(a, ldB(Bb + (size_t)k * DXc + nn * 16, DXc, lane), acc[nn]);
  }
  int row0 = (lane >> 4) << 3, col = lane & 15;
#pragma unroll
  for (int nn = 0; nn < 4; ++nn) {
    float bias = B[(ng << 6) + nn * 16 + col];
    for (int r = 0; r < 8; ++r) {
      int m = mt * 16 + row0 + r;
      size_t idx = (size_t)m * DXc + (ng << 6) + nn * 16 + col;
      pre1[idx] = X[idx] + (acc[nn][r] + bias) * mask[m];
    }
  }
}

// ---------------------------------------------------------------------------
// LayerNorm over rows of 256.
// ---------------------------------------------------------------------------
__global__ __launch_bounds__(256) void ln256_kernel(
    const float* __restrict__ in, const float* __restrict__ g,
    const float* __restrict__ b, float* __restrict__ out) {
  __shared__ float red[256];
  int row = blockIdx.x, t = threadIdx.x;
  float v = in[(size_t)row * DXc + t];
  red[t] = v; __syncthreads();
  for (int s = 128; s > 0; s >>= 1) {
    if (t < s) red[t] += red[t + s];
    __syncthreads();
  }
  float mean = red[0] * (1.f / DXc);
  __syncthreads();
  float d = v - mean;
  red[t] = d * d; __syncthreads();
  for (int s = 128; s > 0; s >>= 1) {
    if (t < s) red[t] += red[t + s];
    __syncthreads();
  }
  float var = red[0] * (1.f / DXc);
  out[(size_t)row * DXc + t] = d * rsqrtf(var + EPSc) * g[t] + b[t];
}

// ---------------------------------------------------------------------------
// Kernel 4a: H = relu(X1 @ linX1_W + b1)  (4-wide N blocking, N=2048)
// ---------------------------------------------------------------------------
__global__ __launch_bounds__(256) void ffx1_kernel(
    const float* __restrict__ X1, const float* __restrict__ W,
    const float* __restrict__ B, float* __restrict__ H) {
  int lane = threadIdx.x & 31;
  int gw = blockIdx.x * (blockDim.x >> 5) + (threadIdx.x >> 5);  // 0..2047
  int mt = gw >> 5, ng = gw & 31;
  v8f acc[4];
  acc[0] = vzero8(); acc[1] = vzero8(); acc[2] = vzero8(); acc[3] = vzero8();
  const float* Ab = X1 + (size_t)mt * 16 * DXc;
  const float* Bb = W + (ng << 6);
  for (int k = 0; k < DXc; k += 4) {
    v2f a = ldA(Ab + k, DXc, lane);
#pragma unroll
    for (int nn = 0; nn < 4; ++nn)
      acc[nn] = wmma4(a, ldB(Bb + (size_t)k * FFXc + nn * 16, FFXc, lane), acc[nn]);
  }
  int row0 = (lane >> 4) << 3, col = lane & 15;
#pragma unroll
  for (int nn = 0; nn < 4; ++nn) {
    float bias = B[(ng << 6) + nn * 16 + col];
    for (int r = 0; r < 8; ++r) {
      int m = mt * 16 + row0 + r;
      H[(size_t)m * FFXc + (ng << 6) + nn * 16 + col] = fmaxf(acc[nn][r] + bias, 0.f);
    }
  }
}

// ---------------------------------------------------------------------------
// Kernel 4b: pre2 = X1 + H @ linX2_W + b2  (4-wide N blocking, K=2048)
// ---------------------------------------------------------------------------
__global__ __launch_bounds__(256) void ffx2_kernel(
    const float* __restrict__ H, const float* __restrict__ W,
    const float* __restrict__ B, const float* __restrict__ X1,
    float* __restrict__ pre2) {
  int lane = threadIdx.x & 31;
  int gw = blockIdx.x * (blockDim.x >> 5) + (threadIdx.x >> 5);  // 0..255
  int mt = gw >> 2, ng = gw & 3;
  v8f acc[4];
  acc[0] = vzero8(); acc[1] = vzero8(); acc[2] = vzero8(); acc[3] = vzero8();
  const float* Ab = H + (size_t)mt * 16 * FFXc;
  const float* Bb = W + (ng << 6);
  for (int k = 0; k < FFXc; k += 4) {
    v2f a = ldA(Ab + k, FFXc, lane);
#pragma unroll
    for (int nn = 0; nn < 4; ++nn)
      acc[nn] = wmma4(a, ldB(Bb + (size_t)k * DXc + nn * 16, DXc, lane), acc[nn]);
  }
  int row0 = (lane >> 4) << 3, col = lane & 15;
#pragma unroll
  for (int nn = 0; nn < 4; ++nn) {
    float bias = B[(ng << 6) + nn * 16 + col];
    for (int r = 0; r < 8; ++r) {
      int m = mt * 16 + row0 + r;
      size_t idx = (size_t)m * DXc + (ng << 6) + nn * 16 + col;
      pre2[idx] = X1[idx] + acc[nn][r] + bias;
    }
  }
}

// ---------------------------------------------------------------------------
// Kernel 5: edge epilogue per 16-edge tile (4 waves)
// ---------------------------------------------------------------------------
__global__ __launch_bounds__(128) void edge_post_kernel(
    const float* __restrict__ E, const float* __restrict__ newE,
    const float* __restrict__ W1, const float* __restrict__ B1,
    const float* __restrict__ W2, const float* __restrict__ B2,
    const float* __restrict__ g1, const float* __restrict__ b1,
    const float* __restrict__ g2, const float* __restrict__ b2,
    float* __restrict__ Eout) {
  __shared__ float Tln[16 * DEc];
  __shared__ float Hs[16 * FFEc];
  __shared__ float T2[16 * DEc];
  int t = threadIdx.x;
  int lane = t & 31, wv = t >> 5;
  size_t base = (size_t)blockIdx.x * 16 * DEc;

  for (int idx = t; idx < 16 * DEc; idx += 128)
    Tln[idx] = E[base + idx] + newE[base + idx];
  __syncthreads();

  for (int rr = wv; rr < 16; rr += 4) {
    float v0 = Tln[rr * DEc + lane], v1 = Tln[rr * DEc + lane + 32];
    float s = v0 + v1;
    for (int o = 16; o > 0; o >>= 1) s += __shfl_xor(s, o, 32);
    float mean = s * (1.f / DEc);
    float d0 = v0 - mean, d1 = v1 - mean;
    float q = d0 * d0 + d1 * d1;
    for (int o = 16; o > 0; o >>= 1) q += __shfl_xor(q, o, 32);
    float inv = rsqrtf(q * (1.f / DEc) + EPSc);
    Tln[rr * DEc + lane]      = d0 * inv * g1[lane]      + b1[lane];
    Tln[rr * DEc + lane + 32] = d1 * inv * g1[lane + 32] + b1[lane + 32];
  }
  __syncthreads();

  int row0 = (lane >> 4) << 3, col = lane & 15;
  for (int nt = wv; nt < 8; nt += 4) {
    v8f acc = vzero8();
    for (int k = 0; k < DEc; k += 4)
      acc = wmma4(ldA(Tln + k, DEc, lane),
                  ldB(W1 + (size_t)k * FFEc + nt * 16, FFEc, lane), acc);
    float bias = B1[nt * 16 + col];
    for (int r = 0; r < 8; ++r)
      Hs[(row0 + r) * FFEc + nt * 16 + col] = fmaxf(acc[r] + bias, 0.f);
  }
  __syncthreads();

  {
    int nt = wv;
    v8f acc = vzero8();
    for (int k = 0; k < FFEc; k += 4)
      acc = wmma4(ldA(Hs + k, FFEc, lane),
                  ldB(W2 + (size_t)k * DEc + nt * 16, DEc, lane), acc);
    float bias = B2[nt * 16 + col];
    for (int r = 0; r < 8; ++r) {
      int off = (row0 + r) * DEc + nt * 16 + col;
      T2[off] = Tln[off] + acc[r] + bias;
    }
  }
  __syncthreads();

  for (int rr = wv; rr < 16; rr += 4) {
    float v0 = T2[rr * DEc + lane], v1 = T2[rr * DEc + lane + 32];
    float s = v0 + v1;
    for (int o = 16; o > 0; o >>= 1) s += __shfl_xor(s, o, 32);
    float mean = s * (1.f / DEc);
    float d0 = v0 - mean, d1 = v1 - mean;
    float q = d0 * d0 + d1 * d1;
    for (int o = 16; o > 0; o >>= 1) q += __shfl_xor(q, o, 32);
    float inv = rsqrtf(q * (1.f / DEc) + EPSc);
    Eout[base + rr * DEc + lane]      = d0 * inv * g2[lane]      + b2[lane];
    Eout[base + rr * DEc + lane + 32] = d1 * inv * g2[lane + 32] + b2[lane + 32];
  }
}

extern "C" void kernel_launch(void* const* d_in, const int* in_sizes, int n_in,
                              void* d_out, int out_size, void* d_ws, size_t ws_size,
                              hipStream_t stream) {
  (void)in_sizes; (void)n_in; (void)out_size; (void)ws_size;
  const float* X     = (const float*)d_in[0];
  const float* E     = (const float*)d_in[1];
  const float* mask  = (const float*)d_in[2];
  const float* qW    = (const float*)d_in[3];
  const float* qB    = (const float*)d_in[4];
  const float* kW    = (const float*)d_in[5];
  const float* kB    = (const float*)d_in[6];
  const float* vW    = (const float*)d_in[7];
  const float* vB    = (const float*)d_in[8];
  const float* emulW = (const float*)d_in[9];
  const float* emulB = (const float*)d_in[10];
  const float* eaddW = (const float*)d_in[11];
  const float* eaddB = (const float*)d_in[12];
  const float* xoutW = (const float*)d_in[13];
  const float* xoutB = (const float*)d_in[14];
  const float* eoutW = (const float*)d_in[15];
  const float* eoutB = (const float*)d_in[16];
  const float* lX1W  = (const float*)d_in[17];
  const float* lX1B  = (const float*)d_in[18];
  const float* lX2W  = (const float*)d_in[19];
  const float* lX2B  = (const float*)d_in[20];
  const float* lE1W  = (const float*)d_in[21];
  const float* lE1B  = (const float*)d_in[22];
  const float* lE2W  = (const float*)d_in[23];
  const float* lE2B  = (const float*)d_in[24];
  const float* nX1g  = (const float*)d_in[25];
  const float* nX1b  = (const float*)d_in[26];
  const float* nX2g  = (const float*)d_in[27];
  const float* nX2b  = (const float*)d_in[28];
  const float* nE1g  = (const float*)d_in[29];
  const float* nE1b  = (const float*)d_in[30];
  const float* nE2g  = (const float*)d_in[31];
  const float* nE2b  = (const float*)d_in[32];

  float* ws   = (float*)d_ws;
  float* Q    = ws;
  float* K    = ws + 262144;
  float* V    = ws + 524288;
  float* wV   = ws + 786432;
  float* pre1 = ws + 1048576;
  float* X1   = ws + 1310720;
  float* pre2 = ws + 1572864;
  float* H    = ws + 1835008;       // 1024*2048
  float* newE = ws + 3932160;       // 16777216

  float* Xout  = (float*)d_out;
  float* EoutP = (float*)d_out + 262144;

  qkv_kernel<<<96, 256, 0, stream>>>(X, qW, qB, kW, kB, vW, vB, mask, Q, K, V);
  edge_fused_kernel<<<1024, 512, 0, stream>>>(E, mask, Q, K, V, emulW, emulB,
                                              eaddW, eaddB, eoutW, eoutB, newE, wV);
  xout_kernel<<<32, 256, 0, stream>>>(wV, xoutW, xoutB, mask, X, pre1);
  ln256_kernel<<<1024, 256, 0, stream>>>(pre1, nX1g, nX1b, X1);
  ffx1_kernel<<<256, 256, 0, stream>>>(X1, lX1W, lX1B, H);
  ffx2_kernel<<<32, 256, 0, stream>>>(H, lX2W, lX2B, X1, pre2);
  ln256_kernel<<<1024, 256, 0, stream>>>(pre2, nX2g, nX2b, Xout);
  edge_post_kernel<<<16384, 128, 0, stream>>>(E, newE, lE1W, lE1B, lE2W, lE2B,
                                              nE1g, nE1b, nE2g, nE2b, EoutP);
}